// GravesStyleNoOffset_14602888807030
// MI455X (gfx1250) — compile-verified
//
#include <hip/hip_runtime.h>
#include <hip/hip_bf16.h>

typedef __attribute__((ext_vector_type(2))) float v2f;
typedef __attribute__((ext_vector_type(8))) float v8f;

#define HIDDEN 512
#define KG     10
#define P3     30          // 3*K_GAUSS
#define BATCH  16
#define TLEN   1024
#define ULEN   600
#define ALPH   80
#define UCHUNK 60          // u-tile per LDS buffer (60*80*4 = 19.2 KB), double-buffered
#define NCHUNK (ULEN / UCHUNK)      // 10
#define CHUNK_F (UCHUNK * ALPH)     // 4800 floats per buffer
#define NWAVE  4           // 128 threads / block
#define PSTRIDE 33         // padded row stride for params transpose (bank-conflict free)
#define POFF_PER_WAVE (16 * PSTRIDE)          // 528 floats per wave
#define CHAR_OFF (NWAVE * POFF_PER_WAVE + 64) // 2176 floats
#define LDS_FLOATS 15424   // max(W stage 15360, CHAR_OFF + 2*4800 = 11776) + pad (61.7 KB)

// Per-lane async 16B copies global -> LDS, tracked by ASYNCcnt.
__device__ __forceinline__ void async_copy_chunk(const float* __restrict__ src,
                                                 const float* dstLds, int tid) {
    const unsigned dbase = (unsigned)(size_t)(const void*)dstLds;  // LDS byte offset
    for (int i = tid; i < CHUNK_F / 4; i += 128) {
        const float* g = src + (size_t)i * 4;
        const unsigned d = dbase + (unsigned)(i * 16);
        asm volatile("global_load_async_to_lds_b128 %0, %1, off"
                     :: "v"(d), "v"(g) : "memory");
    }
}

__global__ __launch_bounds__(128)
void graves_window_kernel(const float* __restrict__ lstm,
                          const float* __restrict__ chars,
                          const float* __restrict__ W,
                          const float* __restrict__ bias,
                          float* __restrict__ out)
{
    __shared__ float lds[LDS_FLOATS];

    const int tid  = threadIdx.x;
    const int wave = tid >> 5;
    const int lane = tid & 31;
    const int m    = lane & 15;        // M row within 16-tile / N col within 16-tile
    const int kb   = (lane >> 4) * 2;  // K sub-pair select per 16x16x4 layout

    const int blk = blockIdx.x;
    const int b   = blk >> 4;          // 16 t-chunks per batch
    const int tch = blk & 15;
    const int t0  = tch * (NWAVE * 16) + wave * 16;   // this wave's 16 t-rows

    // ---------------- Phase 1: stage W[512,30] into LDS ----------------
    {
        const float4* w4 = reinterpret_cast<const float4*>(W);
        float4* dst = reinterpret_cast<float4*>(lds);
        for (int i = tid; i < (HIDDEN * P3) / 4; i += blockDim.x)
            dst[i] = w4[i];
    }
    __syncthreads();

    // ---------------- GEMM1: params[16t x 30] = lstm[16t x 512] @ W ----------------
    v8f acc0 = {};   // p = 0..15
    v8f acc1 = {};   // p = 16..29 (cols 30,31 padded with zero B)
    {
        const float* lrow = lstm + ((size_t)b * TLEN + (t0 + m)) * HIDDEN;
        const int n1 = m + 16;
        const bool n1ok = (n1 < P3);
        for (int h = 0; h < HIDDEN; h += 4) {
            const float2 av2 = *reinterpret_cast<const float2*>(lrow + h + kb);
            v2f a; a.x = av2.x; a.y = av2.y;

            const float* wr0 = lds + (h + kb) * P3;
            const float* wr1 = wr0 + P3;
            v2f b0; b0.x = wr0[m];  b0.y = wr1[m];
            float t1a = wr0[n1 < P3 ? n1 : 0];   // clamp addr; selected away below
            float t1b = wr1[n1 < P3 ? n1 : 0];
            v2f b1; b1.x = n1ok ? t1a : 0.f; b1.y = n1ok ? t1b : 0.f;

            acc0 = __builtin_amdgcn_wmma_f32_16x16x4_f32(false, a, false, b0,
                                                         (short)0, acc0, false, false);
            acc1 = __builtin_amdgcn_wmma_f32_16x16x4_f32(false, a, false, b1,
                                                         (short)0, acc1, false, false);
        }
    }

    // ---------------- Transpose exp(params) through LDS ----------------
    __syncthreads();   // everyone done reading W region; safe to overwrite
    {
        float* pbase = lds + wave * POFF_PER_WAVE;
        const int rbase = (lane >> 4) * 8;   // C/D layout: lanes 0-15 -> M=j, 16-31 -> M=8+j
#pragma unroll
        for (int j = 0; j < 8; ++j) {
            const int r = rbase + j;
            pbase[r * PSTRIDE + m] = __expf(acc0[j] + bias[m]);
            const int p1 = 16 + m;
            if (p1 < P3)
                pbase[r * PSTRIDE + p1] = __expf(acc1[j] + bias[p1]);
        }
    }
    __syncthreads();

    // Each lane grabs mixture params for its t-row (row m of this wave's tile)
    float pa[KG], pb[KG], pk[KG];
    {
        const float* prow = lds + wave * POFF_PER_WAVE + m * PSTRIDE;
#pragma unroll
        for (int k = 0; k < KG; ++k) {
            pa[k] = prow[k];
            pb[k] = prow[KG + k];
            pk[k] = prow[2 * KG + k];
        }
    }

    // ---------------- Fused phi generation + GEMM2, async double-buffered ----------------
    v8f dacc[5];
#pragma unroll
    for (int nt = 0; nt < 5; ++nt) dacc[nt] = (v8f){};

    const float* csrc = chars + (size_t)b * ULEN * ALPH;
    const float* buf0 = lds + CHAR_OFF;
    const float* buf1 = lds + CHAR_OFF + CHUNK_F;

    // Prologue: fill buffer 0 (W region reads are fenced by the barriers above)
    async_copy_chunk(csrc, buf0, tid);
    asm volatile("s_wait_asynccnt 0x0" ::: "memory");
    __syncthreads();

    for (int uc = 0; uc < NCHUNK; ++uc) {
        // Kick off next chunk into the other buffer while we compute this one.
        if (uc + 1 < NCHUNK)
            async_copy_chunk(csrc + (size_t)(uc + 1) * CHUNK_F,
                             ((uc + 1) & 1) ? buf1 : buf0, tid);

        const float* cbuf = (uc & 1) ? buf1 : buf0;
        for (int ut = 0; ut < UCHUNK / 4; ++ut) {
            // phi for A-operand: lane holds (M=m, K=kb) and (M=m, K=kb+1)
            const float u0 = (float)(uc * UCHUNK + ut * 4 + kb);
            const float u1 = u0 + 1.0f;
            float phi0 = 0.f, phi1 = 0.f;
#pragma unroll
            for (int k = 0; k < KG; ++k) {
                const float d0 = pk[k] - u0;
                const float d1 = pk[k] - u1;
                phi0 += pa[k] * __expf(-pb[k] * d0 * d0);
                phi1 += pa[k] * __expf(-pb[k] * d1 * d1);
            }
            v2f av; av.x = phi0; av.y = phi1;

            const float* crow = cbuf + (ut * 4 + kb) * ALPH;
#pragma unroll
            for (int nt = 0; nt < 5; ++nt) {
                v2f bv;
                bv.x = crow[nt * 16 + m];
                bv.y = crow[ALPH + nt * 16 + m];
                dacc[nt] = __builtin_amdgcn_wmma_f32_16x16x4_f32(false, av, false, bv,
                                                                 (short)0, dacc[nt],
                                                                 false, false);
            }
        }

        // My async loads for chunk uc+1 are done; barrier also guarantees every
        // wave finished computing on the buffer that gets refilled next round.
        asm volatile("s_wait_asynccnt 0x0" ::: "memory");
        __syncthreads();
    }

    // ---------------- Epilogue: D tiles -> out[b, t0+r, a] ----------------
    {
        float* orow = out + ((size_t)b * TLEN + t0) * ALPH;
        const int rbase = (lane >> 4) * 8;
#pragma unroll
        for (int nt = 0; nt < 5; ++nt) {
            const int col = nt * 16 + m;
#pragma unroll
            for (int j = 0; j < 8; ++j) {
                orow[(size_t)(rbase + j) * ALPH + col] = dacc[nt][j];
            }
        }
    }
}

extern "C" void kernel_launch(void* const* d_in, const int* in_sizes, int n_in,
                              void* d_out, int out_size, void* d_ws, size_t ws_size,
                              hipStream_t stream) {
    (void)in_sizes; (void)n_in; (void)out_size; (void)d_ws; (void)ws_size;
    const float* lstm  = (const float*)d_in[0];   // [16,1024,512]
    const float* chars = (const float*)d_in[1];   // [16,600,80]
    const float* W     = (const float*)d_in[2];   // [512,30]
    const float* bias  = (const float*)d_in[3];   // [30]
    float* out = (float*)d_out;                   // [16,1024,80]

    dim3 grid(BATCH * 16);   // 16 batches x 16 t-chunks (64 t-rows per block)
    dim3 block(128);         // 4 wave32s, one 16-row t-tile each
    graves_window_kernel<<<grid, block, 0, stream>>>(lstm, chars, W, bias, out);
}